// BetaGNN_1151051236048
// MI455X (gfx1250) — compile-verified
//
#include <hip/hip_runtime.h>
#include <hip/hip_bf16.h>
#include <math.h>

#define HID 256

typedef float v2f __attribute__((ext_vector_type(2)));
typedef float v8f __attribute__((ext_vector_type(8)));

// ---------------------------------------------------------------------------
// K1: h0[i][j] = relu(W_in[j,0]*beta + W_in[j,1]*beta^2 + W_in[j,2]*deg + b_in[j])
// One block (256 threads) per node.
// ---------------------------------------------------------------------------
__global__ void input_layer_kernel(const float* __restrict__ beta,
                                   const float* __restrict__ degree,
                                   const float* __restrict__ W_in,
                                   const float* __restrict__ b_in,
                                   float* __restrict__ H) {
    const int i = blockIdx.x;
    const int j = threadIdx.x;
    const float b = beta[i];
    const float d = degree[i];
    float acc = fmaf(W_in[j * 3 + 0], b, b_in[j]);
    acc = fmaf(W_in[j * 3 + 1], b * b, acc);
    acc = fmaf(W_in[j * 3 + 2], d, acc);
    H[(size_t)i * HID + j] = fmaxf(acc, 0.0f);
}

// ---------------------------------------------------------------------------
// Zero scratch accumulators (graph replays re-run this every call).
// ---------------------------------------------------------------------------
__global__ void zero_kernel(float4* __restrict__ p, size_t n4) {
    size_t i = (size_t)blockIdx.x * blockDim.x + threadIdx.x;
    if (i < n4) p[i] = make_float4(0.f, 0.f, 0.f, 0.f);
}

// ---------------------------------------------------------------------------
// K2/K3: COO SpMM scatter: Y[row] += val * X[col], 256-wide rows.
// One wave32 per edge; lane t handles columns t, t+32, ..., t+224 so both the
// gather loads and the f32 atomics are fully coalesced. X (51 MB) is
// L2-resident (192 MB L2), so the random gathers/atomics stay on-die.
// ---------------------------------------------------------------------------
__global__ void spmm_scatter_kernel(const int* __restrict__ rows,
                                    const int* __restrict__ cols,
                                    const float* __restrict__ vals,
                                    const float* __restrict__ X,
                                    float* __restrict__ Y, int E) {
    const int e = blockIdx.x * 8 + (threadIdx.x >> 5);
    const int lane = threadIdx.x & 31;
    if (e >= E) return;
    const int r = rows[e];
    const int c = cols[e];
    const float v = vals[e];
    const float* __restrict__ x = X + (size_t)c * HID;
    float* __restrict__ y = Y + (size_t)r * HID;
#pragma unroll
    for (int j = 0; j < HID / 32; ++j) {
        const int col = lane + j * 32;
        unsafeAtomicAdd(y + col, v * x[col]);   // global_atomic_add_f32
    }
}

// ---------------------------------------------------------------------------
// K4: D = relu(AH @ W1^T + A2H @ W2^T) with V_WMMA_F32_16X16X4_F32.
//
// One wave computes a 32x64 output panel (2 row tiles x 4 col tiles,
// 8 x v8f accumulators). Each B fragment (W^T 4x16 tile) feeds TWO WMMAs,
// halving W re-read traffic through L2 (~0.8 GB vs 1.6 GB for 16-row panels)
// and raising per-wave intensity to ~10.7 FLOP per L2 byte.
//
// f32 A-tile layout (16x4, MxK): lane l<16 holds A[m][k..k+1], lane l>=16
// holds A[m][k+2..k+3]  -> one contiguous float2 load per lane per k-step.
// B = W^T with W row-major [out,in], so the mirrored B tile load is also a
// contiguous float2 from W's rows. EXEC stays all-ones through every WMMA
// (loads are clamped, only the final stores are predicated).
// ---------------------------------------------------------------------------
__global__ void __launch_bounds__(128)
dual_gemm_relu_kernel(const float* __restrict__ AH,
                      const float* __restrict__ A2H,
                      const float* __restrict__ W1,
                      const float* __restrict__ W2,
                      float* __restrict__ Out, int P) {
    const int lane = threadIdx.x & 31;
    const int wave = threadIdx.x >> 5;       // column group: 0..3
    const int row0 = blockIdx.x * 32;        // 32-row panel per block
    const int c0   = wave * 64;

    const int m  = lane & 15;                // A: row in tile; B: col in tile
    const int kh = (lane >> 4) << 1;         // k sub-offset: 0 or 2

    int r0 = row0 + m;       if (r0 >= P) r0 = P - 1;   // clamp: EXEC all-ones
    int r1 = row0 + 16 + m;  if (r1 >= P) r1 = P - 1;

    const float* __restrict__ a1p0 = AH  + (size_t)r0 * HID + kh;
    const float* __restrict__ a1p1 = AH  + (size_t)r1 * HID + kh;
    const float* __restrict__ a2p0 = A2H + (size_t)r0 * HID + kh;
    const float* __restrict__ a2p1 = A2H + (size_t)r1 * HID + kh;
    const float* __restrict__ w1p  = W1  + (size_t)(c0 + m) * HID + kh;
    const float* __restrict__ w2p  = W2  + (size_t)(c0 + m) * HID + kh;

    v8f accA[4] = {};   // row tile 0, col tiles 0..3
    v8f accB[4] = {};   // row tile 1, col tiles 0..3

    for (int k = 0; k < HID; k += 4) {
        const v2f a10 = *(const v2f*)(a1p0 + k);
        const v2f a11 = *(const v2f*)(a1p1 + k);
        const v2f a20 = *(const v2f*)(a2p0 + k);
        const v2f a21 = *(const v2f*)(a2p1 + k);
#pragma unroll
        for (int nt = 0; nt < 4; ++nt) {
            const v2f b1 = *(const v2f*)(w1p + (size_t)nt * 16 * HID + k);
            const v2f b2 = *(const v2f*)(w2p + (size_t)nt * 16 * HID + k);
            // each B fragment is consumed by two WMMAs (row tiles 0 and 1)
            accA[nt] = __builtin_amdgcn_wmma_f32_16x16x4_f32(false, a10, false, b1, (short)0, accA[nt], false, false);
            accB[nt] = __builtin_amdgcn_wmma_f32_16x16x4_f32(false, a11, false, b1, (short)0, accB[nt], false, false);
            accA[nt] = __builtin_amdgcn_wmma_f32_16x16x4_f32(false, a20, false, b2, (short)0, accA[nt], false, false);
            accB[nt] = __builtin_amdgcn_wmma_f32_16x16x4_f32(false, a21, false, b2, (short)0, accB[nt], false, false);
        }
    }

    // C/D layout: VGPR v, lanes 0-15 -> (M=v, N=lane); lanes 16-31 -> (M=v+8).
    const int rbase = row0 + ((lane >> 4) << 3);
    const int cbase = c0 + m;
#pragma unroll
    for (int nt = 0; nt < 4; ++nt) {
#pragma unroll
        for (int v = 0; v < 8; ++v) {
            const int ra = rbase + v;
            const int rb = rbase + 16 + v;
            if (ra < P) Out[(size_t)ra * HID + cbase + nt * 16] = fmaxf(accA[nt][v], 0.0f);
            if (rb < P) Out[(size_t)rb * HID + cbase + nt * 16] = fmaxf(accB[nt][v], 0.0f);
        }
    }
}

// ---------------------------------------------------------------------------
// K5: g[i] = softplus(dot(h[i], W_out) + b_out). One wave per node.
// ---------------------------------------------------------------------------
__global__ void output_layer_kernel(const float* __restrict__ H,
                                    const float* __restrict__ Wout,
                                    const float* __restrict__ bout,
                                    float* __restrict__ g, int P) {
    const int i = blockIdx.x * 8 + (threadIdx.x >> 5);
    const int lane = threadIdx.x & 31;
    if (i >= P) return;
    const float* __restrict__ h = H + (size_t)i * HID;
    float acc = 0.0f;
#pragma unroll
    for (int j = 0; j < HID / 32; ++j)
        acc = fmaf(h[lane + j * 32], Wout[lane + j * 32], acc);
#pragma unroll
    for (int off = 16; off > 0; off >>= 1)
        acc += __shfl_xor(acc, off, 32);
    if (lane == 0) {
        const float x = acc + bout[0];
        g[i] = (x > 20.0f) ? x : log1pf(expf(x));   // stable softplus
    }
}

// ---------------------------------------------------------------------------
extern "C" void kernel_launch(void* const* d_in, const int* in_sizes, int n_in,
                              void* d_out, int out_size, void* d_ws, size_t ws_size,
                              hipStream_t stream) {
    const float* beta   = (const float*)d_in[0];
    const float* degree = (const float*)d_in[1];
    const int*   A_rows = (const int*)  d_in[2];
    const int*   A_cols = (const int*)  d_in[3];
    const float* A_vals = (const float*)d_in[4];
    const float* W_in   = (const float*)d_in[5];
    const float* b_in   = (const float*)d_in[6];
    const float* W_mp1  = (const float*)d_in[7];
    const float* W_mp2  = (const float*)d_in[8];
    const float* W_out  = (const float*)d_in[9];
    const float* b_out  = (const float*)d_in[10];
    float* out = (float*)d_out;

    const int P = in_sizes[0];   // 50000
    const int E = in_sizes[2];   // 800000

    const size_t matElems = (size_t)P * HID;
    float* H0  = (float*)d_ws;           // h0, later reused for h1
    float* AH  = H0 + matElems;
    float* A2H = AH + matElems;

    // 1) input features + first linear + relu
    input_layer_kernel<<<P, HID, 0, stream>>>(beta, degree, W_in, b_in, H0);

    // 2) zero AH and A2H (contiguous) for the atomic scatters
    const size_t n4 = (2 * matElems) / 4;
    zero_kernel<<<(unsigned)((n4 + 255) / 256), 256, 0, stream>>>((float4*)AH, n4);

    // 3) one-hop and two-hop propagation
    spmm_scatter_kernel<<<(E + 7) / 8, 256, 0, stream>>>(A_rows, A_cols, A_vals, H0, AH, E);
    spmm_scatter_kernel<<<(E + 7) / 8, 256, 0, stream>>>(A_rows, A_cols, A_vals, AH, A2H, E);

    // 4) fused dual GEMM + relu on the matrix cores (f32 WMMA, 32x64 per wave)
    dual_gemm_relu_kernel<<<(P + 31) / 32, 128, 0, stream>>>(AH, A2H, W_mp1, W_mp2, H0, P);

    // 5) output head
    output_layer_kernel<<<(P + 7) / 8, 256, 0, stream>>>(H0, W_out, b_out, out, P);
}